// DeepseekV3TopkRouter_4423816315508
// MI455X (gfx1250) — compile-verified
//
#include <hip/hip_runtime.h>
#include <hip/hip_bf16.h>
#include <stdint.h>

// DeepseekV3 top-k router for gfx1250 (MI455X), wave32.
// One wave per token; lane l owns experts [8l, 8l+8).
// Cross-lane traffic via DPP16 (no LDS pipe) + ballot/readlane.

#define N_EXPERTS       256
#define N_GROUP         8
#define TOPK_GROUP      4
#define TOP_K           8
#define ROUTED_SCALE    2.5f
#define LANES           32
#define EXP_PER_LANE    8                 // 256 / 32
#define WAVES_PER_BLOCK 8
#define BLOCK_THREADS   (WAVES_PER_BLOCK * LANES)

// Map float -> u32 so unsigned compare == float compare (total order;
// negatives map below +0.0). Masked candidates use 0.0f, like the reference.
__device__ __forceinline__ unsigned mono_key(float f) {
  unsigned u = __float_as_uint(f);
  return (u & 0x80000000u) ? ~u : (u | 0x80000000u);
}

#if __has_builtin(__builtin_amdgcn_update_dpp)
#define DSV3_HAVE_DPP 1
template <int CTRL>
__device__ __forceinline__ unsigned dpp_mov_u32(unsigned v) {
  // old = 0, row_mask = 0xF, bank_mask = 0xF, bound_ctrl = 1 (0-fill).
  return (unsigned)__builtin_amdgcn_update_dpp(0, (int)v, CTRL, 0xF, 0xF, true);
}
template <int CTRL>
__device__ __forceinline__ float dpp_mov_f32(float v) {
  return __int_as_float(
      __builtin_amdgcn_update_dpp(0, __float_as_int(v), CTRL, 0xF, 0xF, true));
}
#endif

// Wave-wide max of a u32; result is wave-uniform (SGPR after readlane/s_max).
__device__ __forceinline__ unsigned wave_max_u32(unsigned v) {
#if DSV3_HAVE_DPP
  unsigned t;
  t = dpp_mov_u32<0x111>(v); v = (t > v) ? t : v;   // row_shr:1
  t = dpp_mov_u32<0x112>(v); v = (t > v) ? t : v;   // row_shr:2
  t = dpp_mov_u32<0x114>(v); v = (t > v) ? t : v;   // row_shr:4
  t = dpp_mov_u32<0x118>(v); v = (t > v) ? t : v;   // row_shr:8
  unsigned g0 = (unsigned)__builtin_amdgcn_readlane((int)v, 15);
  unsigned g1 = (unsigned)__builtin_amdgcn_readlane((int)v, 31);
  return (g0 > g1) ? g0 : g1;
#else
#pragma unroll
  for (int m = 16; m >= 1; m >>= 1) {
    unsigned o = (unsigned)__shfl_xor((int)v, m, LANES);
    v = (o > v) ? o : v;
  }
  return (unsigned)__builtin_amdgcn_readfirstlane((int)v);
#endif
}

__device__ __forceinline__ unsigned ballot32(bool p) {
#if __has_builtin(__builtin_amdgcn_ballot_w32)
  return __builtin_amdgcn_ballot_w32(p);
#else
  return (unsigned)__ballot((int)p);
#endif
}

__global__ __launch_bounds__(BLOCK_THREADS, 2) void dsv3_topk_router(
    const float* __restrict__ logits,   // [T, 256]
    const float* __restrict__ bias,     // [256]
    int*   __restrict__ out_idx,        // [T, 8] int32
    float* __restrict__ out_w,          // [T, 8] f32
    int    n_tokens)
{
  __shared__ float smem_scores[WAVES_PER_BLOCK * N_EXPERTS];

  const int lane  = threadIdx.x & (LANES - 1);
  const int wave  = threadIdx.x >> 5;
  const int token = blockIdx.x * WAVES_PER_BLOCK + wave;
  if (token >= n_tokens) return;   // wave-uniform; full wave stays active

  const float* row = logits + (size_t)token * N_EXPERTS;

  // gfx1250 speculative prefetch of this wave's 1KiB row (global_prefetch_b8).
  __builtin_prefetch(row + lane * EXP_PER_LANE, 0, 0);

  // Fully coalesced: wave covers one contiguous 1KiB row, 2 x b128 per lane.
  const float4* row4  = (const float4*)row;
  const float4* bias4 = (const float4*)bias;
  float4 x0 = row4[lane * 2 + 0];
  float4 x1 = row4[lane * 2 + 1];
  float4 b0 = bias4[lane * 2 + 0];
  float4 b1 = bias4[lane * 2 + 1];

  float x[8] = {x0.x, x0.y, x0.z, x0.w, x1.x, x1.y, x1.z, x1.w};
  float b[8] = {b0.x, b0.y, b0.z, b0.w, b1.x, b1.y, b1.z, b1.w};

  float s[8];   // sigmoid(logit)           (weights come from these)
  float c[8];   // sigmoid(logit) + bias    (selection uses these)
#pragma unroll
  for (int j = 0; j < 8; ++j) {
    float e = __expf(-x[j]);                 // v_exp_f32 path
    float d = 1.0f + e;
    float r = __builtin_amdgcn_rcpf(d);      // v_rcp_f32
    r = r * (2.0f - d * r);                  // one Newton step (~0.5 ulp)
    s[j] = r;
    c[j] = r + b[j];
  }

  // Stash uncorrected scores in LDS for the uniform-index weight gather.
  float* my_s = smem_scores + wave * N_EXPERTS;
  ((float4*)my_s)[lane * 2 + 0] = make_float4(s[0], s[1], s[2], s[3]);
  ((float4*)my_s)[lane * 2 + 1] = make_float4(s[4], s[5], s[6], s[7]);

  // ---- group score = top-2 sum of corrected scores within each group of 32.
  // Group g = lanes [4g, 4g+4); per-lane top-2, then quad_perm DPP merges.
  float m1 = -__builtin_inff(), m2 = -__builtin_inff();
#pragma unroll
  for (int j = 0; j < 8; ++j) {
    float v  = c[j];
    float hi = fmaxf(m1, v);
    float lo = fminf(m1, v);
    m1 = hi;
    m2 = fmaxf(m2, lo);
  }
  {
#if DSV3_HAVE_DPP
    float o1 = dpp_mov_f32<0xB1>(m1);   // quad_perm [1,0,3,2]  (xor 1)
    float o2 = dpp_mov_f32<0xB1>(m2);
#else
    float o1 = __shfl_xor(m1, 1, LANES);
    float o2 = __shfl_xor(m2, 1, LANES);
#endif
    float hi = fmaxf(m1, o1), lo = fminf(m1, o1);
    m2 = fmaxf(lo, fmaxf(m2, o2));
    m1 = hi;
#if DSV3_HAVE_DPP
    o1 = dpp_mov_f32<0x4E>(m1);         // quad_perm [2,3,0,1]  (xor 2)
    o2 = dpp_mov_f32<0x4E>(m2);
#else
    o1 = __shfl_xor(m1, 2, LANES);
    o2 = __shfl_xor(m2, 2, LANES);
#endif
    hi = fmaxf(m1, o1); lo = fminf(m1, o1);
    m2 = fmaxf(lo, fmaxf(m2, o2));
    m1 = hi;
  }
  const float gscore = m1 + m2;      // valid in all 4 lanes of the group

  // ---- top-4 groups, tie-break by lower group index (jax top_k semantics).
  const int mygid = lane >> 2;
  int rank = 0;
#pragma unroll
  for (int h = 0; h < N_GROUP; ++h) {
    float gh = __int_as_float(
        __builtin_amdgcn_readlane(__float_as_int(gscore), h * 4));
    rank += ((gh > gscore) || (gh == gscore && h < mygid)) ? 1 : 0;
  }
  const bool gsel = (rank < TOPK_GROUP);

  // ---- candidate keys; masked-out experts participate as exactly 0.0f.
  unsigned key[8];
#pragma unroll
  for (int j = 0; j < 8; ++j) {
    float cand = gsel ? c[j] : 0.0f;
    key[j] = mono_key(cand);
  }

  int sel_i[TOP_K];

#pragma unroll
  for (int r = 0; r < TOP_K; ++r) {
    // Wave-wide argmax with exact lower-index tie-break.
    unsigned lmax = key[0];
#pragma unroll
    for (int j = 1; j < 8; ++j) lmax = (key[j] > lmax) ? key[j] : lmax;
    const unsigned gmax = wave_max_u32(lmax);            // wave-uniform

    // Owner = lowest matching lane (expert ids ascend with lane, so this is
    // already the global lower-index tie-break across lanes).
    const unsigned mm    = ballot32(lmax == gmax);
    const int      owner = (int)__builtin_ctz(mm);

    // Smallest matching slot within each lane (descending scan keeps lowest).
    unsigned cj = 0xFFu;
#pragma unroll
    for (int j = 7; j >= 0; --j) cj = (key[j] == gmax) ? (unsigned)j : cj;

    const int sel = owner * EXP_PER_LANE +
                    __builtin_amdgcn_readlane((int)cj, owner);
    sel_i[r] = sel;

    // Remove the winner (key 0 ranks below everything, incl. masked 0.0).
    const int base = lane * EXP_PER_LANE;
#pragma unroll
    for (int j = 0; j < 8; ++j)
      key[j] = ((base + j) == sel) ? 0u : key[j];
  }

  // Gather the 8 uncorrected scores in one LDS clause (uniform addresses).
  float sel_w[TOP_K];
#pragma unroll
  for (int r = 0; r < TOP_K; ++r) sel_w[r] = my_s[sel_i[r]];

  float wsum = 0.0f;
#pragma unroll
  for (int r = 0; r < TOP_K; ++r) wsum += sel_w[r];

  if (lane == 0) {
    const size_t ob = (size_t)token * TOP_K;
    *(int4*)(out_idx + ob)     = make_int4(sel_i[0], sel_i[1], sel_i[2], sel_i[3]);
    *(int4*)(out_idx + ob + 4) = make_int4(sel_i[4], sel_i[5], sel_i[6], sel_i[7]);
    const float scale = ROUTED_SCALE / (wsum + 1e-20f);
    *(float4*)(out_w + ob)     = make_float4(sel_w[0] * scale, sel_w[1] * scale,
                                             sel_w[2] * scale, sel_w[3] * scale);
    *(float4*)(out_w + ob + 4) = make_float4(sel_w[4] * scale, sel_w[5] * scale,
                                             sel_w[6] * scale, sel_w[7] * scale);
  }
}

extern "C" void kernel_launch(void* const* d_in, const int* in_sizes, int n_in,
                              void* d_out, int out_size, void* d_ws, size_t ws_size,
                              hipStream_t stream) {
  const float* logits = (const float*)d_in[0];   // [T, 256] f32
  const float* bias   = (const float*)d_in[1];   // [256] f32
  const int T = in_sizes[0] / N_EXPERTS;

  // d_out layout: T*8 int32 indices, then T*8 f32 weights (tuple order).
  int*   out_idx = (int*)d_out;
  float* out_w   = (float*)d_out + (size_t)T * TOP_K;

  const int blocks = (T + WAVES_PER_BLOCK - 1) / WAVES_PER_BLOCK;
  dsv3_topk_router<<<blocks, BLOCK_THREADS, 0, stream>>>(logits, bias, out_idx, out_w, T);
}